// SSD_R34_NMS_29394756173835
// MI455X (gfx1250) — compile-verified
//
#include <hip/hip_runtime.h>
#include <hip/hip_bf16.h>

// ---------------- problem constants ----------------
#define BATCH       32
#define NANCH       15130
#define NCLS        81
#define MAXO        200
#define CAP         1024          // per-class candidate cap (>25 sigma above expected ~440)
#define SCORE_TH    0.05f
#define CRIT        0.5f
#define SCALE_XY    0.1f
#define SCALE_WH    0.2f

typedef __attribute__((ext_vector_type(2))) float v2f;
typedef __attribute__((ext_vector_type(8))) float v8f;
typedef unsigned int uint32x4 __attribute__((ext_vector_type(4)));
typedef int          int32x8  __attribute__((ext_vector_type(8)));
typedef int          int32x4  __attribute__((ext_vector_type(4)));

// ---------------- TDM: DMA one contiguous f32 row (global -> LDS) ----------------
#if __has_builtin(__builtin_amdgcn_tensor_load_to_lds)
#define HAVE_TDM 1
__device__ __forceinline__ void tdm_load_row(unsigned ldsAddr, unsigned long long ga, unsigned nElems) {
  // D# group0: count=1, lds_addr, global_addr[56:0], type=2 ("image")
  uint32x4 g0 = { 1u,
                  ldsAddr,
                  (unsigned)(ga & 0xFFFFFFFFull),
                  (unsigned)(((ga >> 32) & 0x01FFFFFFull) | (2u << 30)) };
  // D# group1: data_size=4B, tensor_dim0=nElems, tensor_dim1=1, tile_dim0=nElems,
  //            tile_dim1=1, tensor_dim0_stride=nElems
  int32x8 g1 = { (int)(2u << 16),
                 (int)((nElems & 0xFFFFu) << 16),
                 (int)(((nElems >> 16) & 0xFFFFu) | (1u << 16)),
                 (int)((nElems & 0xFFFFu) << 16),
                 1,
                 (int)nElems,
                 0,
                 0 };
  int32x4 gz = {0, 0, 0, 0};
#if defined(__clang_major__) && (__clang_major__ >= 23)
  int32x8 gz8 = {0, 0, 0, 0, 0, 0, 0, 0};
  __builtin_amdgcn_tensor_load_to_lds(g0, g1, gz, gz, gz8, 0);
#else
  __builtin_amdgcn_tensor_load_to_lds(g0, g1, gz, gz, 0);
#endif
}
#else
#define HAVE_TDM 0
#endif

// =====================================================================
// Kernel 1: box decode + softmax denominator via fp32 WMMA row-reduction
//   grid = (ceil(946/8)=119, 32), block = 256 (8 waves, 16 anchors/wave)
// =====================================================================
__global__ void __launch_bounds__(256)
k_decode_softmax(const float* __restrict__ ploc, const float* __restrict__ plabel,
                 const float* __restrict__ dboxes, float* __restrict__ boxesWs,
                 float* __restrict__ invSum) {
  const int b    = blockIdx.y;
  const int lane = threadIdx.x & 31;
  const int wave = threadIdx.x >> 5;
  const int n0   = (blockIdx.x * 8 + wave) * 16;   // first anchor of this wave
  const int row  = lane & 15;
  int n = n0 + row; if (n > NANCH - 1) n = NANCH - 1;   // clamp so EXEC stays all-1s for WMMA

  const float* pl = plabel + (size_t)b * NCLS * NANCH;
  const int cbase = (lane >> 4) * 2;                // lanes 0-15 -> K0,K1 ; lanes 16-31 -> K2,K3

  v8f acc = {0.f, 0.f, 0.f, 0.f, 0.f, 0.f, 0.f, 0.f};
  v2f ones; ones.x = 1.0f; ones.y = 1.0f;           // B = 4x16 all-ones => D = row sums

  for (int c = 0; c < 84; c += 4) {                 // 21 chunks cover classes 0..80 (pad 81..83)
    const int c0 = c + cbase;
    v2f a;
    a.x = (c0     < NCLS) ? __expf(pl[(size_t)c0       * NANCH + n]) : 0.0f;
    a.y = (c0 + 1 < NCLS) ? __expf(pl[(size_t)(c0 + 1) * NANCH + n]) : 0.0f;
    acc = __builtin_amdgcn_wmma_f32_16x16x4_f32(false, a, false, ones,
                                                (short)0, acc, false, false);
  }

  // D layout: lane0 acc[v] = row v (col 0); lane16 acc[v] = row 8+v
  if (lane == 0 || lane == 16) {
    const int rbase = (lane == 0) ? 0 : 8;
    for (int r = 0; r < 8; ++r) {
      const int nn = n0 + rbase + r;
      if (nn < NANCH) invSum[(size_t)b * NANCH + nn] = 1.0f / acc[r];
    }
  }

  // box decode: lanes 0..15 handle one anchor each
  if (lane < 16) {
    const int nn = n0 + row;
    if (nn < NANCH) {
      const float* pb = ploc + (size_t)b * 4 * NANCH;
      const float px = pb[nn];
      const float py = pb[(size_t)NANCH + nn];
      const float pw = pb[2 * (size_t)NANCH + nn];
      const float ph = pb[3 * (size_t)NANCH + nn];
      const float dcx = dboxes[nn * 4 + 0], dcy = dboxes[nn * 4 + 1];
      const float dw  = dboxes[nn * 4 + 2], dh  = dboxes[nn * 4 + 3];
      const float cx = px * SCALE_XY * dw + dcx;
      const float cy = py * SCALE_XY * dh + dcy;
      const float w  = __expf(pw * SCALE_WH) * dw;
      const float h  = __expf(ph * SCALE_WH) * dh;
      float* o = boxesWs + ((size_t)b * NANCH + nn) * 4;
      o[0] = cx - 0.5f * w; o[1] = cy - 0.5f * h;
      o[2] = cx + 0.5f * w; o[3] = cy + 0.5f * h;
    }
  }
}

// ---------------- IoU on ltrb boxes ----------------
__device__ __forceinline__ float iou_ltrb(const float* a, const float* c) {
  const float lx = fmaxf(a[0], c[0]), ly = fmaxf(a[1], c[1]);
  const float rx = fminf(a[2], c[2]), ry = fminf(a[3], c[3]);
  const float w = fmaxf(rx - lx, 0.0f), h = fmaxf(ry - ly, 0.0f);
  const float inter = w * h;
  const float aa = (a[2] - a[0]) * (a[3] - a[1]);
  const float ab = (c[2] - c[0]) * (c[3] - c[1]);
  return inter / (aa + ab - inter + 1e-12f);
}

// =====================================================================
// Kernel 2: per-(image,class) threshold -> top-200 -> greedy NMS
//   grid = (80, 32), block = 256. Class row staged to LDS via TDM.
// =====================================================================
__global__ void __launch_bounds__(256)
k_perclass_nms(const float* __restrict__ plabel, const float* __restrict__ invSum,
               const float* __restrict__ boxesWs, float* __restrict__ valsWs,
               int* __restrict__ anchorWs) {
  __shared__ float sLogits[NANCH];
  __shared__ float sScore[CAP];
  __shared__ int   sIdx[CAP];
  __shared__ int   sCnt[256];
  __shared__ float sBx[MAXO * 4];
  __shared__ int   sKeep[MAXO];

  const int tid = threadIdx.x;
  const int b   = blockIdx.y;
  const int cls = blockIdx.x + 1;                   // skip background class 0
  const float* rowp = plabel + ((size_t)b * NCLS + cls) * NANCH;

#if HAVE_TDM
  if (tid < 32) {                                   // wave 0 owns the DMA + its TENSORcnt
    tdm_load_row((unsigned)(size_t)(void*)sLogits,
                 (unsigned long long)(size_t)(const void*)rowp, NANCH);
    __builtin_amdgcn_s_wait_tensorcnt(0);
  }
#else
  for (int n = tid; n < NANCH; n += 256) sLogits[n] = rowp[n];
#endif
  __syncthreads();

  const float* inv = invSum + (size_t)b * NANCH;

  // pass 1: count candidates per thread (deterministic compaction, no atomics)
  int my = 0;
  for (int n = tid; n < NANCH; n += 256)
    if (__expf(sLogits[n]) * inv[n] > SCORE_TH) my++;
  sCnt[tid] = my;
  __syncthreads();
  for (int off = 1; off < 256; off <<= 1) {         // Hillis-Steele inclusive scan
    const int v = (tid >= off) ? sCnt[tid - off] : 0;
    __syncthreads();
    sCnt[tid] += v;
    __syncthreads();
  }
  const int total = sCnt[255];
  int pos = sCnt[tid] - my;

  // pass 2: write candidates + pad tail
  for (int n = tid; n < NANCH; n += 256) {
    const float sc = __expf(sLogits[n]) * inv[n];
    if (sc > SCORE_TH) {
      if (pos < CAP) { sScore[pos] = sc; sIdx[pos] = n; }
      pos++;
    }
  }
  for (int i = tid; i < CAP; i += 256)
    if (i >= total) { sScore[i] = -1.0f; sIdx[i] = 0; }
  __syncthreads();

  // bitonic sort: score descending, anchor-index ascending on ties (top_k stability)
  for (int k = 2; k <= CAP; k <<= 1) {
    for (int j = k >> 1; j > 0; j >>= 1) {
      for (int i = tid; i < CAP; i += 256) {
        const int ixj = i ^ j;
        if (ixj > i) {
          const float s1 = sScore[i], s2 = sScore[ixj];
          const int   i1 = sIdx[i],   i2 = sIdx[ixj];
          const bool before = (s1 > s2) || (s1 == s2 && i1 < i2);
          const bool dir    = ((i & k) == 0);
          if (before != dir) {
            sScore[i] = s2; sScore[ixj] = s1;
            sIdx[i]   = i2; sIdx[ixj]   = i1;
          }
        }
      }
      __syncthreads();
    }
  }

  // gather boxes of the top-200 candidates
  if (tid < MAXO) {
    const int a = sIdx[tid];
    const float* bp = boxesWs + ((size_t)b * NANCH + a) * 4;
    sBx[tid * 4 + 0] = bp[0]; sBx[tid * 4 + 1] = bp[1];
    sBx[tid * 4 + 2] = bp[2]; sBx[tid * 4 + 3] = bp[3];
    sKeep[tid] = 1;
  }

  // greedy NMS: 200 sequential rounds, parallel suppression
  for (int i = 0; i < MAXO; ++i) {
    __syncthreads();
    if (tid > i && tid < MAXO && sKeep[tid] && sKeep[i] && sScore[i] > 0.0f) {
      if (iou_ltrb(&sBx[i * 4], &sBx[tid * 4]) >= CRIT) sKeep[tid] = 0;
    }
  }
  __syncthreads();

  if (tid < MAXO) {
    const float v = (sKeep[tid] && sScore[tid] > 0.0f) ? sScore[tid] : 0.0f;
    const size_t o = ((size_t)b * (NCLS - 1) + (cls - 1)) * MAXO + tid;
    valsWs[o]   = v;
    anchorWs[o] = sIdx[tid];
  }
}

// =====================================================================
// Kernel 3: per-image top-200 over 80*200 candidates (stable argmax loop)
//   grid = 32, block = 256
// =====================================================================
__global__ void __launch_bounds__(256)
k_final_topk(const float* __restrict__ boxesWs, float* __restrict__ valsWs,
             const int* __restrict__ anchorWs, float* __restrict__ out) {
  const int TOT = (NCLS - 1) * MAXO;                // 16000
  __shared__ float rv[256];
  __shared__ int   ri[256];
  const int tid = threadIdx.x;
  const int b   = blockIdx.x;
  float* vals = valsWs + (size_t)b * TOT;

  for (int k = 0; k < MAXO; ++k) {
    float bv = -1e30f; int bi = 0;
    for (int f = tid; f < TOT; f += 256) {
      const float v = vals[f];
      if (v > bv) { bv = v; bi = f; }               // strict '>' keeps lowest index on ties
    }
    rv[tid] = bv; ri[tid] = bi;
    __syncthreads();
    for (int off = 128; off > 0; off >>= 1) {
      if (tid < off) {
        const float v2 = rv[tid + off]; const int i2 = ri[tid + off];
        if (v2 > rv[tid] || (v2 == rv[tid] && i2 < ri[tid])) { rv[tid] = v2; ri[tid] = i2; }
      }
      __syncthreads();
    }
    if (tid == 0) {
      const int f = ri[0];
      float v = rv[0]; if (v < 0.0f) v = 0.0f;
      const int cls    = f / MAXO + 1;
      const int anchor = anchorWs[(size_t)b * TOT + f];
      const float* bx  = boxesWs + ((size_t)b * NANCH + anchor) * 4;
      float* ob = out + ((size_t)b * MAXO + k) * 4;
      ob[0] = bx[0]; ob[1] = bx[1]; ob[2] = bx[2]; ob[3] = bx[3];
      out[(size_t)BATCH * MAXO * 4 + (size_t)b * MAXO + k]                       = (float)cls;
      out[(size_t)BATCH * MAXO * 4 + (size_t)BATCH * MAXO + (size_t)b * MAXO + k] = v;
      vals[f] = -1e30f;                             // consume
    }
    __syncthreads();
  }
}

// =====================================================================
extern "C" void kernel_launch(void* const* d_in, const int* in_sizes, int n_in,
                              void* d_out, int out_size, void* d_ws, size_t ws_size,
                              hipStream_t stream) {
  const float* ploc   = (const float*)d_in[0];   // [32,4,15130]
  const float* plabel = (const float*)d_in[1];   // [32,81,15130]
  const float* dboxes = (const float*)d_in[2];   // [15130,4]
  float* out = (float*)d_out;                    // boxes | labels | scores

  float* ws       = (float*)d_ws;
  float* boxesWs  = ws;                                          // 32*15130*4
  float* invSum   = ws + (size_t)BATCH * NANCH * 4;              // 32*15130
  float* valsWs   = invSum + (size_t)BATCH * NANCH;              // 32*80*200
  int*   anchorWs = (int*)(valsWs + (size_t)BATCH * (NCLS - 1) * MAXO);

  const int wavesPerImg  = (NANCH + 15) / 16;                    // 946
  const int blocksPerImg = (wavesPerImg + 7) / 8;                // 119
  dim3 g1(blocksPerImg, BATCH);
  k_decode_softmax<<<g1, 256, 0, stream>>>(ploc, plabel, dboxes, boxesWs, invSum);

  dim3 g2(NCLS - 1, BATCH);
  k_perclass_nms<<<g2, 256, 0, stream>>>(plabel, invSum, boxesWs, valsWs, anchorWs);

  k_final_topk<<<BATCH, 256, 0, stream>>>(boxesWs, valsWs, anchorWs, out);
}